// VectorQuantizer_52012053954796
// MI455X (gfx1250) — compile-verified
//
#include <hip/hip_runtime.h>
#include <hip/hip_bf16.h>

// CDNA5 / gfx1250 VQ-VAE vector quantizer.
// N = 65536 rows, D = 256, K = 4096 codes.
// Fused GEMM (f32 WMMA 16x16x4) + argmin + gather + loss.
// A tile staged with GLOBAL_LOAD_ASYNC_TO_LDS_B128 (ASYNCcnt path);
// B tiles staged via clause-batched b128 loads + LDS transpose;
// WMMA fragments preloaded per 64-deep block to avoid per-pair dscnt stalls.

typedef __attribute__((ext_vector_type(2))) float v2f;
typedef __attribute__((ext_vector_type(8))) float v8f;

#define VQ_N 65536
#define VQ_D 256
#define VQ_K 4096
#define ROWS_PER_BLK 32
#define NBLK (VQ_N / ROWS_PER_BLK)   // 2048
#define STRA 260                     // padded LDS stride for A (bank-conflict-free b64)
#define STRB 68                      // padded LDS stride for B (code-major)

// ---- CDNA5 async global->LDS copy helpers ---------------------------------
__device__ __forceinline__ void async_copy_b128(void* lds_dst, const void* gsrc) {
    // generic shared pointer: addr[31:0] is the LDS byte offset (aperture layout)
    asm volatile("global_load_async_to_lds_b128 %0, %1, off"
                 :: "v"((unsigned)(size_t)lds_dst),
                    "v"((unsigned long long)(size_t)gsrc)
                 : "memory");
}

__device__ __forceinline__ void wait_asynccnt0() {
#if __has_builtin(__builtin_amdgcn_s_wait_asynccnt)
    __builtin_amdgcn_s_wait_asynccnt(0);
#else
    asm volatile("s_wait_asynccnt 0x0" ::: "memory");
#endif
}

// ---------------------------------------------------------------------------
// Kernel 1: per-code squared norms  esq[k] = sum_d E[d,k]^2
// ---------------------------------------------------------------------------
__global__ __launch_bounds__(256) void vq_emb_norms(const float* __restrict__ E,
                                                    float* __restrict__ esq) {
    int k = blockIdx.x * 256 + threadIdx.x;   // 4096 total, coalesced over k
    float s = 0.0f;
    #pragma unroll 8
    for (int d = 0; d < VQ_D; ++d) {
        float e = E[(long long)d * VQ_K + k];
        s += e * e;
    }
    esq[k] = s;
}

// ---------------------------------------------------------------------------
// Kernel 2: main fused GEMM + argmin + gather + per-block loss partial
// ---------------------------------------------------------------------------
__global__ __launch_bounds__(256) void vq_main(const float* __restrict__ z,
                                               const float* __restrict__ E,
                                               const float* __restrict__ esq,
                                               float* __restrict__ out,
                                               float* __restrict__ blockSums) {
    __shared__ __align__(16) float Alds[ROWS_PER_BLK * STRA]; // 33,280 B
    __shared__ __align__(16) float Blds[64 * STRB];           // 17,408 B (reused for reductions)
    __shared__ int rowIdx[ROWS_PER_BLK];

    const int tid    = threadIdx.x;
    const int lane   = tid & 31;          // wave32
    const int wave   = tid >> 5;          // 8 waves
    const int l16    = lane & 15;
    const int lhalf  = lane >> 4;         // 0: lanes 0-15, 1: lanes 16-31
    const int msub   = wave & 1;          // M-subtile: rows [0..15] or [16..31]
    const int cchunk = wave >> 1;         // which 16-code chunk (0..3)
    const long long rowBase = (long long)blockIdx.x * ROWS_PER_BLK;

    // ---- stage A tile (32 rows x 256 f32) via async global->LDS, no VGPR hop
    #pragma unroll
    for (int it = 0; it < 8; ++it) {
        const int i  = tid + it * 256;    // float4 index, 2048 total
        const int r  = i >> 6;
        const int c4 = i & 63;
        async_copy_b128(&Alds[r * STRA + c4 * 4],
                        z + (rowBase + r) * VQ_D + c4 * 4);
    }
    wait_asynccnt0();   // own writes done; cross-wave visibility via barrier below

    float bestS[8];
    int   bestI[8];
    #pragma unroll
    for (int v = 0; v < 8; ++v) { bestS[v] = 3.4e38f; bestI[v] = 0; }

    // ---- K loop: 64 codes per iteration (4 chunks of 16 across wave pairs) ----
    for (int cb = 0; cb < VQ_K; cb += 64) {
        v8f acc = {0.f, 0.f, 0.f, 0.f, 0.f, 0.f, 0.f, 0.f};

        for (int dblk = 0; dblk < 4; ++dblk) {        // 4 x 64 reduction dims
            const int d0 = dblk * 64;
            __syncthreads();   // previous readers of Blds done (also covers A-stage)

            // stage B sub-tile E[d0..d0+63][cb..cb+63] code-major into LDS:
            // batch all 4 b128 loads first (one clause), then transpose-store.
            float4 tv[4];
            #pragma unroll
            for (int it = 0; it < 4; ++it) {
                const int i  = tid + it * 256;   // (d, code-quad), 1024 total
                const int d  = i >> 4;
                const int c4 = i & 15;
                tv[it] = *(const float4*)(E + (long long)(d0 + d) * VQ_K + cb + c4 * 4);
            }
            #pragma unroll
            for (int it = 0; it < 4; ++it) {
                const int i  = tid + it * 256;
                const int d  = i >> 4;
                const int c4 = i & 15;
                Blds[(c4 * 4 + 0) * STRB + d] = tv[it].x;
                Blds[(c4 * 4 + 1) * STRB + d] = tv[it].y;
                Blds[(c4 * 4 + 2) * STRB + d] = tv[it].z;
                Blds[(c4 * 4 + 3) * STRB + d] = tv[it].w;
            }
            __syncthreads();

            // Preload all fragments of this 64-deep block into registers
            // (32 x ds_load_2addr_b64-pairable loads in flight, single wait),
            // then run 16 back-to-back accumulator-chained f32 WMMAs.
            const float* aBase = &Alds[(msub * 16 + l16) * STRA + d0 + 2 * lhalf];
            const float* bBase = &Blds[(cchunk * 16 + l16) * STRB + 2 * lhalf];
            v2f af[16], bf[16];
            #pragma unroll
            for (int s = 0; s < 16; ++s) {
                af[s] = *(const v2f*)(aBase + s * 4);   // A 16x4 fragment
                bf[s] = *(const v2f*)(bBase + s * 4);   // B 4x16 fragment
            }
            #pragma unroll
            for (int s = 0; s < 16; ++s) {
                acc = __builtin_amdgcn_wmma_f32_16x16x4_f32(
                        /*neg_a=*/false, af[s], /*neg_b=*/false, bf[s],
                        /*c_mod=*/(short)0, acc, /*reuse_a=*/false, /*reuse_b=*/false);
            }
        }

        // score = ||e||^2 - 2 x.e   (||x||^2 irrelevant to argmin)
        const int   code = cb + cchunk * 16 + l16;   // N-dim depends on lane&15 only
        const float e2   = esq[code];
        #pragma unroll
        for (int v = 0; v < 8; ++v) {
            float s = e2 - 2.0f * acc[v];
            if (s < bestS[v]) { bestS[v] = s; bestI[v] = code; }
        }
    }

    // ---- per-row argmin across 16 lanes x 4 code-chunk waves ----
    __syncthreads();                       // done reading Blds; reuse as scratch
    float* redS = Blds;                    // 2048 floats
    int*   redI = (int*)(Blds + 2048);     // 2048 ints (fits in 4352-float Blds)
    #pragma unroll
    for (int v = 0; v < 8; ++v) {
        int row  = msub * 16 + v + 8 * lhalf;   // C/D layout: VGPR v -> M = v (+8 hi half)
        int slot = cchunk * 16 + l16;
        redS[row * 64 + slot] = bestS[v];
        redI[row * 64 + slot] = bestI[v];
    }
    __syncthreads();
    if (tid < ROWS_PER_BLK) {
        float bs = redS[tid * 64];
        int   bi = redI[tid * 64];
        for (int j = 1; j < 64; ++j) {
            float s = redS[tid * 64 + j];
            if (s < bs) { bs = s; bi = redI[tid * 64 + j]; }
        }
        rowIdx[tid] = bi;
    }
    __syncthreads();

    // ---- gather winning codes, write output, accumulate (q - z)^2 ----
    float partial = 0.0f;
    for (int i = tid; i < ROWS_PER_BLK * VQ_D; i += 256) {
        int r = i >> 8;
        int d = i & 255;
        int idx = rowIdx[r];
        float q  = E[(long long)d * VQ_K + idx];   // strided gather, L2-resident
        float zz = Alds[r * STRA + d];
        out[(rowBase + r) * VQ_D + d] = q;         // quantized_st == q numerically
        float df = q - zz;
        partial += df * df;
    }

    // deterministic block-level loss reduction
    __syncthreads();
    redS[tid] = partial;
    __syncthreads();
    for (int off = 128; off > 0; off >>= 1) {
        if (tid < off) redS[tid] += redS[tid + off];
        __syncthreads();
    }
    if (tid == 0) blockSums[blockIdx.x] = redS[0];
}

// ---------------------------------------------------------------------------
// Kernel 3: deterministic final loss reduction, scale by 1.25 / (N*D)
// ---------------------------------------------------------------------------
__global__ __launch_bounds__(256) void vq_finalize(const float* __restrict__ blockSums,
                                                   float* __restrict__ lossOut) {
    __shared__ float s[256];
    float p = 0.0f;
    for (int i = threadIdx.x; i < NBLK; i += 256) p += blockSums[i];
    s[threadIdx.x] = p;
    __syncthreads();
    for (int off = 128; off > 0; off >>= 1) {
        if (threadIdx.x < off) s[threadIdx.x] += s[threadIdx.x + off];
        __syncthreads();
    }
    if (threadIdx.x == 0)
        *lossOut = s[0] * 1.25f / (float)((long long)VQ_N * VQ_D);
}

// ---------------------------------------------------------------------------
extern "C" void kernel_launch(void* const* d_in, const int* in_sizes, int n_in,
                              void* d_out, int out_size, void* d_ws, size_t ws_size,
                              hipStream_t stream) {
    const float* z = (const float*)d_in[0];   // [64,32,32,256] f32 -> [65536,256]
    const float* E = (const float*)d_in[1];   // [256,4096] f32
    float* out = (float*)d_out;               // [65536*256] quantized + [1] loss

    float* esq       = (float*)d_ws;                  // 4096 floats
    float* blockSums = (float*)d_ws + VQ_K;           // 2048 floats
    float* lossOut   = out + (long long)VQ_N * VQ_D;  // last element of d_out

    vq_emb_norms<<<VQ_K / 256, 256, 0, stream>>>(E, esq);
    vq_main<<<NBLK, 256, 0, stream>>>(z, E, esq, out, blockSums);
    vq_finalize<<<1, 256, 0, stream>>>(blockSums, lossOut);
}